// SCDGN_57535381897866
// MI455X (gfx1250) — compile-verified
//
#include <hip/hip_runtime.h>
#include <hip/hip_bf16.h>

// ---------------------------------------------------------------------------
// SCDGN pipeline for MI455X (gfx1250, wave32).
// Big GEMMs: V_WMMA_F32_16X16X4_F32, 128x128 block tile, 32x64 wave tile,
// double-buffered LDS, all fragment loads as aligned ds_load_b64 (stride-34
// padded LDS, conflict-free). Tiles staged by the Tensor Data Mover where the
// layout allows (tensor_load_to_lds + s_wait_tensorcnt), per-thread float4
// staging otherwise. SpMM: float4 gathers + global_atomic_add_f32.
// ---------------------------------------------------------------------------

#define NN_   4096
#define FEAT_ 500
#define HID_  512
#define NE_   65536
#define TSTEPS_ 8

typedef __attribute__((ext_vector_type(2))) float        v2f;
typedef __attribute__((ext_vector_type(8))) float        v8f;
typedef __attribute__((ext_vector_type(4))) unsigned int v4u;
typedef __attribute__((ext_vector_type(4))) int          v4i;
typedef __attribute__((ext_vector_type(8))) int          v8i;

#if defined(__has_builtin)
#  if __has_builtin(__builtin_amdgcn_tensor_load_to_lds) && \
      __has_builtin(__builtin_amdgcn_s_wait_tensorcnt)
#    define HAVE_TDM 1
#  endif
#endif
#ifndef HAVE_TDM
#  define HAVE_TDM 0
#endif

#if HAVE_TDM
#warning SCDGN_TDM_PATH_ENABLED_tensor_load_to_lds
#else
#warning SCDGN_TDM_PATH_DISABLED_using_perthread_staging
#endif

#define BM 128
#define BN 128
#define BK 32
#define SP (BK + 2)   // LDS row stride: 136B -> 8B-aligned b64 frags, 16 banks

#if HAVE_TDM
// 2D tile load via Tensor Data Mover: tile_rows x tile_cols 4-byte elements
// from a row-major tensor (row stride `row_stride` elements), written to LDS
// with 2 DWORDs of padding after every 32 DWORDs (=> LDS row stride 34).
__device__ __forceinline__ void scdgn_tdm_load_2d(
    unsigned lds_off, const float* gptr,
    unsigned tile_cols, unsigned tile_rows,
    unsigned tensor_cols, unsigned tensor_rows,
    unsigned row_stride)
{
    unsigned long long ga = (unsigned long long)(uintptr_t)gptr;
    v4u g0;
    g0[0] = 1u;                                            // count=1 (valid), user mode
    g0[1] = lds_off;                                       // lds_addr (bytes)
    g0[2] = (unsigned)(ga & 0xffffffffu);                  // global_addr[31:0]
    g0[3] = (unsigned)((ga >> 32) & 0x01ffffffu)           // global_addr[56:32]
          | 0x80000000u;                                   // type=2 ("image")
    v8i g1;
    g1[0] = (int)((2u << 16)                               // data_size = 4B
                | (1u << 20)                               // pad_enable
                | (4u << 22)                               // pad_interval = 32 DW
                | (1u << 25));                             // pad_amount   = 2 DW
    g1[1] = (int)((tensor_cols & 0xffffu) << 16);          // tensor_dim0[15:0]
    g1[2] = (int)((tensor_cols >> 16) | ((tensor_rows & 0xffffu) << 16));
    g1[3] = (int)((tensor_rows >> 16) | (tile_cols << 16)); // tile_dim0
    g1[4] = (int)tile_rows;                                 // tile_dim1 (tile_dim2=0)
    g1[5] = (int)row_stride;                                // tensor_dim0_stride[31:0]
    g1[6] = 0;
    g1[7] = 0;
    v4i z4 = {0, 0, 0, 0};
#if __clang_major__ >= 23
    v8i z8 = {0, 0, 0, 0, 0, 0, 0, 0};
    __builtin_amdgcn_tensor_load_to_lds(g0, g1, z4, z4, z8, 0);
#else
    __builtin_amdgcn_tensor_load_to_lds(g0, g1, z4, z4, 0);
#endif
}
#endif

// ----------------------------- WMMA GEMM -----------------------------------
// C[M,Nn] = A[M,K] @ op(B) (+ bias); op(B)=B [K,Nn] or B^T with B [Nn,K].
// 256 threads = 8 waves (4 M x 2 N), wave tile 32x64 = 2x4 WMMA accumulators.
// LDS: As[m][k] and Bs[n][k], both k-fast with stride 34.
__global__ __launch_bounds__(256)
void scdgn_gemm_f32_wmma(const float* __restrict__ A, const float* __restrict__ B,
                         const float* __restrict__ bias, float* __restrict__ C,
                         int M, int Nn, int K, int transB)
{
    __shared__ float As[2][BM * SP];
    __shared__ float Bs[2][BN * SP];

    const int tid  = threadIdx.x;
    const int lane = tid & 31;
    const int wave = tid >> 5;          // 0..7
    const int wm   = wave >> 1;         // 0..3 -> M offset wm*32
    const int wn   = wave & 1;          // 0..1 -> N offset wn*64
    const int hi   = lane >> 4;
    const int l16  = lane & 15;
    const int blockM = blockIdx.y * BM;
    const int blockN = blockIdx.x * BN;
    const int m0 = wm * 32;
    const int n0 = wn * 64;

    v8f acc[2][4] = {};
    float4 aReg[4], bReg[4];

    // ---------------- staging helpers (double-buffered) ----------------
    auto loadA = [&](int kt) {
        #pragma unroll
        for (int i = 0; i < 4; ++i) {
            int idx = tid + i * 256;
            int r   = idx >> 3;                    // 0..127
            int cq  = (idx & 7) * 4;               // 0..28
            aReg[i] = *reinterpret_cast<const float4*>(
                A + (size_t)(blockM + r) * K + kt + cq);
        }
    };
    auto storeA = [&](int buf) {
        #pragma unroll
        for (int i = 0; i < 4; ++i) {
            int idx = tid + i * 256;
            int r   = idx >> 3;
            int cq  = (idx & 7) * 4;
            float* p = &As[buf][r * SP + cq];
            p[0] = aReg[i].x; p[1] = aReg[i].y; p[2] = aReg[i].z; p[3] = aReg[i].w;
        }
    };
    auto loadB = [&](int kt) {
        if (!transB) {
            // B[K,Nn]: float4 along n, will transpose-scatter into Bs[n][k]
            #pragma unroll
            for (int i = 0; i < 4; ++i) {
                int idx = tid + i * 256;
                int r   = idx >> 5;                // k: 0..31
                int cq  = (idx & 31) * 4;          // n quad: 0..124
                bReg[i] = *reinterpret_cast<const float4*>(
                    B + (size_t)(kt + r) * Nn + blockN + cq);
            }
        } else {
            // B[Nn,K]: float4 along k, rows land contiguously in Bs[n][k]
            #pragma unroll
            for (int i = 0; i < 4; ++i) {
                int idx = tid + i * 256;
                int cc  = idx >> 3;                // n: 0..127
                int rq  = (idx & 7) * 4;           // k quad: 0..28
                bReg[i] = *reinterpret_cast<const float4*>(
                    B + (size_t)(blockN + cc) * K + kt + rq);
            }
        }
    };
    auto storeB = [&](int buf) {
        if (!transB) {
            #pragma unroll
            for (int i = 0; i < 4; ++i) {
                int idx = tid + i * 256;
                int r   = idx >> 5;
                int cq  = (idx & 31) * 4;
                float* p = &Bs[buf][0];
                p[(cq + 0) * SP + r] = bReg[i].x;
                p[(cq + 1) * SP + r] = bReg[i].y;
                p[(cq + 2) * SP + r] = bReg[i].z;
                p[(cq + 3) * SP + r] = bReg[i].w;
            }
        } else {
            #pragma unroll
            for (int i = 0; i < 4; ++i) {
                int idx = tid + i * 256;
                int cc  = idx >> 3;
                int rq  = (idx & 7) * 4;
                float* p = &Bs[buf][cc * SP + rq];
                p[0] = bReg[i].x; p[1] = bReg[i].y; p[2] = bReg[i].z; p[3] = bReg[i].w;
            }
        }
    };

    auto stageIssue = [&](int kt, int buf) {
#if HAVE_TDM
        if (wave == 0) {
            scdgn_tdm_load_2d((unsigned)(uintptr_t)&As[buf][0],
                              A + (size_t)blockM * K + kt,
                              BK, BM, (unsigned)(K - kt), (unsigned)(M - blockM),
                              (unsigned)K);
            if (transB)
                scdgn_tdm_load_2d((unsigned)(uintptr_t)&Bs[buf][0],
                                  B + (size_t)blockN * K + kt,
                                  BK, BN, (unsigned)(K - kt), (unsigned)(Nn - blockN),
                                  (unsigned)K);
        }
        if (!transB) loadB(kt);
#else
        loadA(kt);
        loadB(kt);
#endif
    };
    auto stageCommit = [&](int buf) {
#if HAVE_TDM
        if (!transB) storeB(buf);
#else
        storeA(buf);
        storeB(buf);
#endif
    };

    // ---------------- prologue ----------------
    stageIssue(0, 0);
    stageCommit(0);
#if HAVE_TDM
    __builtin_amdgcn_s_wait_tensorcnt(0);
#endif
    __syncthreads();

    // ---------------- main K loop (double-buffered) ----------------
    const int nTiles = K / BK;
    for (int t = 0; t < nTiles; ++t) {
        const int cur = t & 1;
        const bool hasNext = (t + 1 < nTiles);
        if (hasNext) stageIssue((t + 1) * BK, cur ^ 1);

        const float* asb = &As[cur][0];
        const float* bsb = &Bs[cur][0];
        #pragma unroll
        for (int kk = 0; kk < BK; kk += 4) {
            const int ka = kk + 2 * hi;            // f32 WMMA K-pair layout
            v2f a[2], b[4];
            #pragma unroll
            for (int i = 0; i < 2; ++i)
                a[i] = *reinterpret_cast<const v2f*>(&asb[(m0 + i * 16 + l16) * SP + ka]);
            #pragma unroll
            for (int j = 0; j < 4; ++j)
                b[j] = *reinterpret_cast<const v2f*>(&bsb[(n0 + j * 16 + l16) * SP + ka]);
            #pragma unroll
            for (int i = 0; i < 2; ++i)
                #pragma unroll
                for (int j = 0; j < 4; ++j)
                    acc[i][j] = __builtin_amdgcn_wmma_f32_16x16x4_f32(
                        false, a[i], false, b[j], (short)0, acc[i][j], false, false);
        }

        if (hasNext) stageCommit(cur ^ 1);
#if HAVE_TDM
        __builtin_amdgcn_s_wait_tensorcnt(0);
#endif
        __syncthreads();
    }

    // ---------------- epilogue: store C (+bias) ----------------
    #pragma unroll
    for (int i = 0; i < 2; ++i) {
        const int bm0 = blockM + m0 + i * 16;
        #pragma unroll
        for (int j = 0; j < 4; ++j) {
            const int bn0 = blockN + n0 + j * 16;
            const float bv = bias ? bias[bn0 + l16] : 0.0f;
            #pragma unroll
            for (int v = 0; v < 8; ++v)
                C[(size_t)(bm0 + v + 8 * hi) * Nn + bn0 + l16] = acc[i][j][v] + bv;
        }
    }
}

// ----------------------- pad knn [N,500] -> [N,512] -------------------------
__global__ void scdgn_pad_knn(const float* __restrict__ knn, float* __restrict__ knnp)
{
    int i   = blockIdx.x * 256 + threadIdx.x;     // over N*512
    int row = i >> 9;
    int col = i & 511;
    knnp[i] = (col < FEAT_) ? knn[(size_t)row * FEAT_ + col] : 0.0f;
}

// ----------------------- row L2-normalize (in place) ------------------------
__global__ __launch_bounds__(256)
void scdgn_rownorm(float* __restrict__ m, int cols)
{
    __shared__ float red[8];
    const int row = blockIdx.x;
    float s = 0.f;
    for (int c = threadIdx.x; c < cols; c += 256) {
        float v = m[(size_t)row * cols + c];
        s += v * v;
    }
    #pragma unroll
    for (int off = 16; off > 0; off >>= 1) s += __shfl_down(s, off);
    if ((threadIdx.x & 31) == 0) red[threadIdx.x >> 5] = s;
    __syncthreads();
    if (threadIdx.x == 0) {
        float t = 0.f;
        #pragma unroll
        for (int w = 0; w < 8; ++w) t += red[w];
        red[0] = 1.0f / fmaxf(sqrtf(t), 1e-12f);
    }
    __syncthreads();
    const float inv = red[0];
    for (int c = threadIdx.x; c < cols; c += 256)
        m[(size_t)row * cols + c] *= inv;
}

// ------------------- ODE init: x = h, ax = 0, stats = 0 ---------------------
__global__ void scdgn_ode_init(const float* __restrict__ h, float* __restrict__ x,
                               float* __restrict__ ax, float* __restrict__ stats)
{
    int i = blockIdx.x * 256 + threadIdx.x;       // over N*HID
    x[i]  = h[i];
    ax[i] = 0.0f;
    if (i < 514) stats[i] = 0.0f;                 // colsum[512] + {sum,sumsq}
}

// --------- SpMM: ax[row] += w * x[col] (one block per edge, float4) ---------
__global__ __launch_bounds__(128)
void scdgn_spmm(const int* __restrict__ ei, const float* __restrict__ ew,
                const float* __restrict__ x, float* __restrict__ ax)
{
    const int e   = blockIdx.x;
    const int row = ei[e];
    const int col = ei[NE_ + e];
    const float w = ew[e];
    const int j   = threadIdx.x * 4;              // 0..508
    const float4 v = *reinterpret_cast<const float4*>(x + (size_t)col * HID_ + j);
    float* dst = ax + (size_t)row * HID_ + j;
    unsafeAtomicAdd(dst + 0, w * v.x);
    unsafeAtomicAdd(dst + 1, w * v.y);
    unsafeAtomicAdd(dst + 2, w * v.z);
    unsafeAtomicAdd(dst + 3, w * v.w);
}

// ------- Euler step: x += alpha*(ax - x) + beta*x0 ; ax = 0 for next --------
__global__ void scdgn_ode_update(float* __restrict__ x, float* __restrict__ ax,
                                 const float* __restrict__ x0,
                                 const float* __restrict__ alpha_p,
                                 const float* __restrict__ beta_p)
{
    int i = blockIdx.x * 256 + threadIdx.x;
    const float alpha = 1.0f / (1.0f + __expf(-alpha_p[0]));
    const float beta  = beta_p[0];
    float xv = x[i];
    x[i]  = xv + alpha * (ax[i] - xv) + beta * x0[i];
    ax[i] = 0.0f;
}

// ----------------- z = relu(nf * z_ode + h) -> out --------------------------
__global__ void scdgn_relu_res(const float* __restrict__ x, const float* __restrict__ h,
                               const float* __restrict__ nf, float* __restrict__ z)
{
    int i = blockIdx.x * 256 + threadIdx.x;
    z[i] = fmaxf(nf[0] * x[i] + h[i], 0.0f);
}

// --------- per-column sums + global sum/sumsq (block = 512 thr, 8 rows) -----
__global__ __launch_bounds__(512)
void scdgn_stats(const float* __restrict__ z, float* __restrict__ stats)
{
    __shared__ float rs[16], rss[16];
    const int col  = threadIdx.x;                 // 0..511
    const int row0 = blockIdx.x * 8;
    float s = 0.f, ss = 0.f;
    #pragma unroll
    for (int r = 0; r < 8; ++r) {
        float v = z[(size_t)(row0 + r) * HID_ + col];
        s += v; ss += v * v;
    }
    unsafeAtomicAdd(&stats[col], s);              // column sum
    float gs = s, gss = ss;
    #pragma unroll
    for (int off = 16; off > 0; off >>= 1) { gs += __shfl_down(gs, off); gss += __shfl_down(gss, off); }
    if ((threadIdx.x & 31) == 0) { rs[threadIdx.x >> 5] = gs; rss[threadIdx.x >> 5] = gss; }
    __syncthreads();
    if (threadIdx.x == 0) {
        float a = 0.f, b = 0.f;
        #pragma unroll
        for (int w = 0; w < 16; ++w) { a += rs[w]; b += rss[w]; }
        unsafeAtomicAdd(&stats[512], a);
        unsafeAtomicAdd(&stats[513], b);
    }
}

// --------- z = (z - colmean) / std(z, ddof=1) (in place on out) -------------
__global__ void scdgn_finalize(float* __restrict__ z, const float* __restrict__ stats)
{
    int i = blockIdx.x * 256 + threadIdx.x;
    const int col = i & (HID_ - 1);
    const float Mtot = (float)NN_ * (float)HID_;
    const float mean = stats[512] / Mtot;
    const float var  = (stats[513] - Mtot * mean * mean) / (Mtot - 1.0f);
    const float inv  = rsqrtf(var);
    const float cm   = stats[col] / (float)NN_;
    z[i] = (z[i] - cm) * inv;
}

// ---------------------------------------------------------------------------
extern "C" void kernel_launch(void* const* d_in, const int* in_sizes, int n_in,
                              void* d_out, int out_size, void* d_ws, size_t ws_size,
                              hipStream_t stream)
{
    const float* knn   = (const float*)d_in[0];   // [N,500]
    const float* adj   = (const float*)d_in[1];   // [N,N]
    const float* nf    = (const float*)d_in[2];   // scalar
    const float* w1    = (const float*)d_in[3];   // [N,N]
    const float* w2    = (const float*)d_in[4];   // [N,N]
    const float* W0    = (const float*)d_in[5];   // [HID,N]
    const float* b0    = (const float*)d_in[6];   // [HID]
    const float* W1    = (const float*)d_in[7];   // [HID,HID]
    const float* b1    = (const float*)d_in[8];   // [HID]
    const float* ew    = (const float*)d_in[9];   // [E]
    const float* alp   = (const float*)d_in[10];  // scalar
    const float* bet   = (const float*)d_in[11];  // scalar
    const int*   ei    = (const int*)d_in[12];    // [2,E] (jax default int32)

    float* out = (float*)d_out;                   // [N,HID]

    // workspace layout
    float* bufA  = (float*)d_ws;                          // N*N
    float* bufB  = bufA + (size_t)NN_ * NN_;              // N*N
    float* stats = bufB + (size_t)NN_ * NN_;              // 514 floats

    float* knnp = bufB;                                   // [N,512]
    float* Hm   = bufB + (size_t)NN_ * HID_;              // [N,512]
    float* hb   = bufB;                                   // [N,512] (reuses knnp)
    float* h2   = bufB + (size_t)NN_ * HID_;              // [N,512] (reuses Hm)
    float* xb   = bufB + 2 * (size_t)NN_ * HID_;          // [N,512]
    float* axb  = bufB + 3 * (size_t)NN_ * HID_;          // [N,512]

    const dim3 blk256(256);
    const int elemBlocks = (NN_ * HID_) / 256;            // 8192

    auto gemm = [&](const float* A, const float* B, const float* bias, float* C,
                    int M, int Nn, int K, int transB) {
        dim3 grid(Nn / BN, M / BM);
        scdgn_gemm_f32_wmma<<<grid, blk256, 0, stream>>>(A, B, bias, C, M, Nn, K, transB);
    };

    // ---- AGCN chain ----
    gemm(adj,  w1,  nullptr, bufA, NN_, NN_, NN_, 0);     // T1 = adj @ w1
    gemm(bufA, adj, nullptr, bufB, NN_, NN_, NN_, 1);     // H1 = T1 @ adj^T
    gemm(bufB, w2,  nullptr, bufA, NN_, NN_, NN_, 0);     // T2 = H1 @ w2
    scdgn_pad_knn<<<elemBlocks, blk256, 0, stream>>>(knn, knnp);
    gemm(bufA, knnp, nullptr, Hm, NN_, HID_, NN_, 0);     // H  = T2 @ knnp  [N,512]
    gemm(Hm,   Hm,  nullptr, bufA, NN_, NN_, HID_, 1);    // embed = H @ H^T
    scdgn_rownorm<<<NN_, blk256, 0, stream>>>(bufA, NN_); // F.normalize(dim=1)

    // ---- extractor MLP ----
    gemm(bufA, W0, b0, hb, NN_, HID_, NN_, 1);            // h  = embed @ W0^T + b0
    gemm(hb,   W1, b1, h2, NN_, HID_, HID_, 1);           // h2 = h @ W1^T + b1

    // ---- ODE diffusion (explicit Euler, T=8) ----
    scdgn_ode_init<<<elemBlocks, blk256, 0, stream>>>(h2, xb, axb, stats);
    for (int t = 0; t < TSTEPS_; ++t) {
        scdgn_spmm<<<NE_, dim3(128), 0, stream>>>(ei, ew, xb, axb);
        scdgn_ode_update<<<elemBlocks, blk256, 0, stream>>>(xb, axb, h2, alp, bet);
    }

    // ---- epilogue: relu residual, standardize ----
    scdgn_relu_res<<<elemBlocks, blk256, 0, stream>>>(xb, h2, nf, out);
    scdgn_stats<<<NN_ / 8, dim3(512), 0, stream>>>(out, stats);
    scdgn_finalize<<<elemBlocks, blk256, 0, stream>>>(out, stats);
}